// GraphSAGEImputer_50766513439458
// MI455X (gfx1250) — compile-verified
//
#include <hip/hip_runtime.h>
#include <hip/hip_bf16.h>
#include <stdint.h>

typedef __attribute__((ext_vector_type(2))) float v2f;
typedef __attribute__((ext_vector_type(8))) float v8f;

#define DIM   128
#define KDIM  256
#define EPS_WSUM 1e-12f
#define EPS_NORM 1e-12f

// ---------------------------------------------------------------------------
// Scatter: one wave per edge. agg[src] += x[dst]*w ; wsum[src] += w
// Each lane handles 4 consecutive features (32 lanes * 4 = 128).
// unsafeAtomicAdd lowers to native global_atomic_add_f32 (no CAS loop).
// ---------------------------------------------------------------------------
__global__ __launch_bounds__(256) void sage_scatter_kernel(
    const float* __restrict__ x,          // [N,128]
    const long long* __restrict__ ei,     // [2,E] int64
    const float* __restrict__ ew,         // [E]
    float* __restrict__ agg,              // [N,128]
    float* __restrict__ wsum,             // [N]
    int E)
{
    int edge = (int)((blockIdx.x * blockDim.x + threadIdx.x) >> 5);
    int lane = threadIdx.x & 31;
    if (edge >= E) return;

    int src = (int)ei[edge];                 // uniform per wave -> scalar load
    int dst = (int)ei[(size_t)E + edge];
    float w = ew[edge];

    const float4* xr = (const float4*)(x + (size_t)dst * DIM);
    float4 v = xr[lane];

    float* ar = agg + (size_t)src * DIM + (lane << 2);
    unsafeAtomicAdd(ar + 0, v.x * w);
    unsafeAtomicAdd(ar + 1, v.y * w);
    unsafeAtomicAdd(ar + 2, v.z * w);
    unsafeAtomicAdd(ar + 3, v.w * w);
    if (lane == 0) unsafeAtomicAdd(wsum + src, w);
}

// ---------------------------------------------------------------------------
// Fused SAGE linear layer: out = relu([x | agg/max(wsum,eps)] @ W^T + b)
// Block = 256 threads = 8 waves. Block owns 16 nodes; wave w owns output
// columns [16w, 16w+16). K-loop of v_wmma_f32_16x16x4_f32.
// ---------------------------------------------------------------------------
__global__ __launch_bounds__(256) void sage_gemm_relu_kernel(
    const float* __restrict__ xin,   // [N,128] layer input
    const float* __restrict__ agg,   // [N,128] weighted neighbor sums
    const float* __restrict__ wsum,  // [N]
    const float* __restrict__ W,     // [128,256] row-major (out, in)
    const float* __restrict__ bias,  // [128]
    float* __restrict__ out,         // [N,128]
    int N)
{
    __shared__ float tile[16][KDIM + 4];   // 16 nodes x 256 (x | neigh), padded

    const int m0 = blockIdx.x * 16;
    const int t  = threadIdx.x;

    // Cooperative LDS fill: iteration r loads row r (node m0+r), column t.
    // Columns 0..127 from xin, 128..255 from agg scaled by 1/max(wsum,eps).
    for (int r = 0; r < 16; ++r) {
        int node = m0 + r;
        float val = 0.0f;
        if (node < N) {
            if (t < DIM) {
                val = xin[(size_t)node * DIM + t];
            } else {
                float inv = 1.0f / fmaxf(wsum[node], EPS_WSUM);
                val = agg[(size_t)node * DIM + (t - DIM)] * inv;
            }
        }
        tile[r][t] = val;
    }
    __syncthreads();

    const int waveId = t >> 5;
    const int lane   = t & 31;
    const int hi     = lane >> 4;     // half-wave select
    const int lo     = lane & 15;
    const int o0     = waveId * 16;   // this wave's output-column tile

    // A (16x4 f32) lane layout: lanes 0-15 hold K=k,k+1 of row M=lo;
    // lanes 16-31 hold K=k+2,k+3 of row M=lo. B (4x16) mirrors it:
    // lane holds W^T[k+2hi .. k+2hi+1][o0+lo] = W[o0+lo][k+2hi ..].
    const float* arow = &tile[lo][2 * hi];
    const float* wrow = W + (size_t)(o0 + lo) * KDIM + 2 * hi;

    v8f c = {};
    #pragma unroll 8
    for (int k = 0; k < KDIM; k += 4) {
        v2f a = *(const v2f*)(arow + k);
        v2f b = *(const v2f*)(wrow + k);
        c = __builtin_amdgcn_wmma_f32_16x16x4_f32(
                /*neg_a=*/false, a, /*neg_b=*/false, b,
                /*c_mod=*/(short)0, c, /*reuse_a=*/false, /*reuse_b=*/false);
    }

    // C/D layout: VGPR v, lanes 0-15 -> (M=v, Ncol=lane); lanes 16-31 -> (M=v+8, Ncol=lane-16)
    const float bb = bias[o0 + lo];
    float res[8];
    #pragma unroll
    for (int v = 0; v < 8; ++v)
        res[v] = fmaxf(c[v] + bb, 0.0f);

    // One uniform branch instead of 8 exec-mask toggles: only the final block
    // takes the guarded path.
    float* po = out + (size_t)(m0 + 8 * hi) * DIM + o0 + lo;
    if (m0 + 16 <= N) {
        #pragma unroll
        for (int v = 0; v < 8; ++v)
            po[(size_t)v * DIM] = res[v];
    } else {
        #pragma unroll
        for (int v = 0; v < 8; ++v)
            if (m0 + 8 * hi + v < N)
                po[(size_t)v * DIM] = res[v];
    }
}

// ---------------------------------------------------------------------------
// Row L2-normalize in place: one wave per node, 4 floats per lane.
// ---------------------------------------------------------------------------
__global__ __launch_bounds__(256) void row_l2norm_kernel(float* __restrict__ h, int N)
{
    int node = (int)((blockIdx.x * blockDim.x + threadIdx.x) >> 5);
    int lane = threadIdx.x & 31;
    if (node >= N) return;

    float4* row = (float4*)(h + (size_t)node * DIM);
    float4 v = row[lane];
    float ss = v.x * v.x + v.y * v.y + v.z * v.z + v.w * v.w;
    #pragma unroll
    for (int off = 16; off > 0; off >>= 1)
        ss += __shfl_xor(ss, off, 32);
    float scale = 1.0f / fmaxf(sqrtf(ss), EPS_NORM);
    v.x *= scale; v.y *= scale; v.z *= scale; v.w *= scale;
    row[lane] = v;
}

// ---------------------------------------------------------------------------
extern "C" void kernel_launch(void* const* d_in, const int* in_sizes, int n_in,
                              void* d_out, int out_size, void* d_ws, size_t ws_size,
                              hipStream_t stream) {
    const float*     x  = (const float*)d_in[0];       // [N,128]
    const long long* ei = (const long long*)d_in[1];   // [2,E] int64
    const float*     ew = (const float*)d_in[2];       // [E]
    const float*     W1 = (const float*)d_in[3];       // [128,256]
    const float*     b1 = (const float*)d_in[4];       // [128]
    const float*     W2 = (const float*)d_in[5];       // [128,256]
    const float*     b2 = (const float*)d_in[6];       // [128]
    float* out = (float*)d_out;                        // [N,128]

    const int N = in_sizes[0] / DIM;
    const int E = in_sizes[2];

    // Workspace layout: agg [N*128] | wsum [N] | h1 [N*128]
    float* agg  = (float*)d_ws;
    float* wsum = agg + (size_t)N * DIM;
    float* h1   = wsum + N;

    const int scatterBlocks = (E + 7) / 8;     // 8 edges (waves) per 256-thread block
    const int gemmBlocks    = (N + 15) / 16;   // 16 nodes per block
    const int normBlocks    = (N + 7) / 8;     // 8 nodes per block

    // ---- Layer 1 ----
    hipMemsetAsync(agg, 0, (size_t)N * (DIM + 1) * sizeof(float), stream); // agg+wsum contiguous
    sage_scatter_kernel<<<scatterBlocks, 256, 0, stream>>>(x, ei, ew, agg, wsum, E);
    sage_gemm_relu_kernel<<<gemmBlocks, 256, 0, stream>>>(x, agg, wsum, W1, b1, h1, N);

    // ---- Layer 2 ----
    hipMemsetAsync(agg, 0, (size_t)N * (DIM + 1) * sizeof(float), stream);
    sage_scatter_kernel<<<scatterBlocks, 256, 0, stream>>>(h1, ei, ew, agg, wsum, E);
    sage_gemm_relu_kernel<<<gemmBlocks, 256, 0, stream>>>(h1, agg, wsum, W2, b2, out, N);

    // ---- Final row L2 normalize ----
    row_l2norm_kernel<<<normBlocks, 256, 0, stream>>>(out, N);
}